// SemiSoftTriHard_41025527611554
// MI455X (gfx1250) — compile-verified
//
#include <hip/hip_runtime.h>
#include <math.h>

typedef __attribute__((ext_vector_type(2))) float v2f;
typedef __attribute__((ext_vector_type(8))) float v8f;
typedef __attribute__((ext_vector_type(4))) int v4i;

#define N_EMB 4096
#define D_EMB 1024
#define K_SEL 64       // sqrt(N)
#define BM 128
#define BN 128
#define BK 16
#define LDS_STRIDE 20  // 16 + 4 pad; 20 dwords = 80B => float4 stores stay 16B aligned
#define THREADS 256
#define NBINS 512
#define MAXC 2048

#define AS_GLOBAL __attribute__((address_space(1)))
#define AS_LOCAL  __attribute__((address_space(3)))

#if defined(__gfx1250__) && __has_builtin(__builtin_amdgcn_global_load_async_to_lds_b128)
#define HAVE_ASYNC_LDS 1
#else
#define HAVE_ASYNC_LDS 0
#endif

#if __has_builtin(__builtin_amdgcn_s_wait_asynccnt)
#define WAIT_ASYNCCNT(n) __builtin_amdgcn_s_wait_asynccnt(n)
#else
#define WAIT_ASYNCCNT(n) asm volatile("s_wait_asynccnt %0" ::"i"(n))
#endif

// ---------------------------------------------------------------------------
// GEMM: out[i][j] = 2 - 2 * dot(A[i,:], B[j,:])   (A,B row-major [N_EMB, D_EMB])
// 128x128 block tile, 8 waves, each wave owns a 32x64 sub-tile = 2x4 WMMA tiles.
// fp32 WMMA (V_WMMA_F32_16X16X4_F32) to preserve reference fp32 top-k selection.
// Double-buffered LDS fed by GLOBAL_LOAD_ASYNC_TO_LDS_B128 (ASYNCcnt-tracked),
// falling back to synchronous load+ds_store if the builtin is unavailable.
// ---------------------------------------------------------------------------
__global__ __launch_bounds__(THREADS)
void gemm_dist_kernel(const float* __restrict__ A, const float* __restrict__ B,
                      float* __restrict__ out) {
#if HAVE_ASYNC_LDS
  __shared__ float As[2][BM * LDS_STRIDE];
  __shared__ float Bs[2][BN * LDS_STRIDE];
#else
  __shared__ float As[1][BM * LDS_STRIDE];
  __shared__ float Bs[1][BN * LDS_STRIDE];
#endif

  const int tid  = threadIdx.x;
  const int lane = tid & 31;
  const int wave = tid >> 5;
  const int wm   = wave >> 1;        // 0..3 : 32-row band
  const int wn   = wave & 1;         // 0..1 : 64-col band
  const int row0 = blockIdx.y * BM;
  const int col0 = blockIdx.x * BN;

  // cooperative load mapping: thread -> (row, 8-float half)
  const int lr = tid >> 1;           // 0..127
  const int lq = (tid & 1) * 8;      // 0 or 8

  // WMMA f32 16x16x4 A/B fragment lane mapping:
  // lanes 0-15 hold K=0(.x)/K=1(.y), lanes 16-31 hold K=2(.x)/K=3(.y)
  const int m16 = lane & 15;
  const int kh  = (lane >> 4) << 1;  // 0 or 2

  v8f acc[2][4] = {};

#if HAVE_ASYNC_LDS
  const float* gaBase = A + (size_t)(row0 + lr) * D_EMB + lq;
  const float* gbBase = B + (size_t)(col0 + lr) * D_EMB + lq;

  auto issue = [&](int kb, int buf) {
    AS_GLOBAL v4i* ga = (AS_GLOBAL v4i*)(gaBase + kb);
    AS_GLOBAL v4i* gb = (AS_GLOBAL v4i*)(gbBase + kb);
    AS_LOCAL v4i* la = (AS_LOCAL v4i*)&As[buf][lr * LDS_STRIDE + lq];
    AS_LOCAL v4i* lb = (AS_LOCAL v4i*)&Bs[buf][lr * LDS_STRIDE + lq];
    // INST_OFFSET applies to both global and LDS addresses (ISA 15.18):
    // two b128s cover this thread's 32B slice of the tile.
    __builtin_amdgcn_global_load_async_to_lds_b128(ga, la, 0, 0);
    __builtin_amdgcn_global_load_async_to_lds_b128(ga, la, 16, 0);
    __builtin_amdgcn_global_load_async_to_lds_b128(gb, lb, 0, 0);
    __builtin_amdgcn_global_load_async_to_lds_b128(gb, lb, 16, 0);
  };

  issue(0, 0);
  int buf = 0;
  for (int kb = 0; kb < D_EMB; kb += BK) {
    if (kb + BK < D_EMB) {
      issue(kb + BK, buf ^ 1);  // prefetch next tile into the other buffer
      WAIT_ASYNCCNT(4);         // this wave's 4 loads for `buf` have landed
    } else {
      WAIT_ASYNCCNT(0);
    }
    __syncthreads();            // all waves' async loads for `buf` visible

#pragma unroll
    for (int kk = 0; kk < 4; ++kk) {
      v2f afrag[2], bfrag[4];
#pragma unroll
      for (int mi = 0; mi < 2; ++mi)
        afrag[mi] = *(const v2f*)&As[buf][(wm * 32 + mi * 16 + m16) * LDS_STRIDE + kk * 4 + kh];
#pragma unroll
      for (int ni = 0; ni < 4; ++ni)
        bfrag[ni] = *(const v2f*)&Bs[buf][(wn * 64 + ni * 16 + m16) * LDS_STRIDE + kk * 4 + kh];
#pragma unroll
      for (int mi = 0; mi < 2; ++mi)
#pragma unroll
        for (int ni = 0; ni < 4; ++ni)
          acc[mi][ni] = __builtin_amdgcn_wmma_f32_16x16x4_f32(
              false, afrag[mi], false, bfrag[ni], (short)0, acc[mi][ni], false, false);
    }
    __syncthreads();            // done reading `buf` before it is refilled
    buf ^= 1;
  }
#else
  for (int kb = 0; kb < D_EMB; kb += BK) {
    const float4* ag = (const float4*)(A + (size_t)(row0 + lr) * D_EMB + kb + lq);
    const float4* bg = (const float4*)(B + (size_t)(col0 + lr) * D_EMB + kb + lq);
    float4 a0 = ag[0], a1 = ag[1];
    float4 b0 = bg[0], b1 = bg[1];
    *(float4*)&As[0][lr * LDS_STRIDE + lq + 0] = a0;
    *(float4*)&As[0][lr * LDS_STRIDE + lq + 4] = a1;
    *(float4*)&Bs[0][lr * LDS_STRIDE + lq + 0] = b0;
    *(float4*)&Bs[0][lr * LDS_STRIDE + lq + 4] = b1;
    __syncthreads();

#pragma unroll
    for (int kk = 0; kk < 4; ++kk) {
      v2f afrag[2], bfrag[4];
#pragma unroll
      for (int mi = 0; mi < 2; ++mi)
        afrag[mi] = *(const v2f*)&As[0][(wm * 32 + mi * 16 + m16) * LDS_STRIDE + kk * 4 + kh];
#pragma unroll
      for (int ni = 0; ni < 4; ++ni)
        bfrag[ni] = *(const v2f*)&Bs[0][(wn * 64 + ni * 16 + m16) * LDS_STRIDE + kk * 4 + kh];
#pragma unroll
      for (int mi = 0; mi < 2; ++mi)
#pragma unroll
        for (int ni = 0; ni < 4; ++ni)
          acc[mi][ni] = __builtin_amdgcn_wmma_f32_16x16x4_f32(
              false, afrag[mi], false, bfrag[ni], (short)0, acc[mi][ni], false, false);
    }
    __syncthreads();
  }
#endif

  // C/D layout: lane n=lane&15 is column; VGPR v holds M=v (lanes 0-15) / M=v+8 (lanes 16-31)
  const int mbase = row0 + wm * 32 + ((lane >> 4) << 3);
  const int nbase = col0 + wn * 64 + (lane & 15);
#pragma unroll
  for (int mi = 0; mi < 2; ++mi) {
#pragma unroll
    for (int v = 0; v < 8; ++v) {
      const int row = mbase + mi * 16 + v;
#pragma unroll
      for (int ni = 0; ni < 4; ++ni)
        out[(size_t)row * N_EMB + nbase + ni * 16] = 2.0f - 2.0f * acc[mi][ni][v];
    }
  }
}

// ---------------------------------------------------------------------------
// Per-row: select the K_SEL smallest off-diagonal distances via histogram
// select, accumulate softplus(beta * (pos - d)) over them.
// ---------------------------------------------------------------------------
__device__ __forceinline__ float softplusf(float z) {
  // log1p(exp(z)), numerically stable
  return fmaxf(z, 0.0f) + log1pf(__expf(-fabsf(z)));
}

__global__ __launch_bounds__(256)
void topk_softmargin_kernel(const float* __restrict__ dist,
                            const int* __restrict__ loss_weight,
                            float* __restrict__ accum) {
  __shared__ unsigned hist[NBINS];
  __shared__ float cand[MAXC];
  __shared__ float red[256];
  __shared__ int redi[256];
  __shared__ unsigned ccount;
  __shared__ int bstar_s, below_s;

  const int tid = threadIdx.x;
  const int r   = blockIdx.x;
  const float beta = (float)(*loss_weight);
  const float* rowp = dist + (size_t)r * N_EMB;
  const float pos = rowp[r];

  for (int b = tid; b < NBINS; b += 256) hist[b] = 0u;
  if (tid == 0) ccount = 0u;
  __syncthreads();

  // pass 1: histogram over [0,4)  (bin = d * NBINS/4)
  for (int j = tid; j < N_EMB; j += 256) {
    if (j == r) continue;
    float d = rowp[j];
    int b = (int)(d * 128.0f);
    b = b < 0 ? 0 : (b > NBINS - 1 ? NBINS - 1 : b);
    atomicAdd(&hist[b], 1u);
  }
  __syncthreads();

  if (tid == 0) {
    unsigned cum = 0;
    int bs = NBINS - 1, bel = 0;
    for (int b = 0; b < NBINS; ++b) {
      unsigned c = hist[b];
      if (cum + c >= (unsigned)K_SEL) { bs = b; bel = (int)cum; break; }
      cum += c;
    }
    bstar_s = bs;
    below_s = bel;
  }
  __syncthreads();
  const int bstar = bstar_s;
  const int below = below_s;

  // pass 2: sum strictly-below bins; collect boundary-bin candidates
  float acc = 0.0f;
  for (int j = tid; j < N_EMB; j += 256) {
    if (j == r) continue;
    float d = rowp[j];
    int b = (int)(d * 128.0f);
    b = b < 0 ? 0 : (b > NBINS - 1 ? NBINS - 1 : b);
    if (b < bstar) {
      acc += softplusf(beta * (pos - d));
    } else if (b == bstar) {
      unsigned idx = atomicAdd(&ccount, 1u);
      if (idx < MAXC) cand[idx] = d;
    }
  }
  __syncthreads();

  int nc = (int)ccount;
  if (nc > MAXC) nc = MAXC;
  int need = K_SEL - below;
  if (need > nc) need = nc;
  if (need < 0) need = 0;

  // extract the `need` smallest boundary-bin candidates
  for (int t = 0; t < need; ++t) {
    float mv = 3.0e38f;
    int mi = -1;
    for (int j = tid; j < nc; j += 256) {
      float v = cand[j];
      if (v < mv) { mv = v; mi = j; }
    }
    red[tid] = mv;
    redi[tid] = mi;
    __syncthreads();
    for (int s = 128; s > 0; s >>= 1) {
      if (tid < s && red[tid + s] < red[tid]) {
        red[tid] = red[tid + s];
        redi[tid] = redi[tid + s];
      }
      __syncthreads();
    }
    if (tid == 0) {
      acc += softplusf(beta * (pos - red[0]));
      if (redi[0] >= 0) cand[redi[0]] = 3.0e38f;
    }
    __syncthreads();
  }

  // block-reduce and accumulate
  red[tid] = acc;
  __syncthreads();
  for (int s = 128; s > 0; s >>= 1) {
    if (tid < s) red[tid] += red[tid + s];
    __syncthreads();
  }
  if (tid == 0) atomicAdd(accum, red[0]);
}

__global__ void init_accum_kernel(float* __restrict__ acc) {
  acc[0] = 0.0f;
  acc[1] = 0.0f;
}

__global__ void finalize_kernel(const float* __restrict__ acc, float* __restrict__ out) {
  out[0] = (acc[0] + acc[1]) / (2.0f * (float)N_EMB * (float)K_SEL);
}

// ---------------------------------------------------------------------------
extern "C" void kernel_launch(void* const* d_in, const int* in_sizes, int n_in,
                              void* d_out, int out_size, void* d_ws, size_t ws_size,
                              hipStream_t stream) {
  (void)in_sizes; (void)n_in; (void)out_size; (void)ws_size;
  const float* grd  = (const float*)d_in[0];   // [N, D]
  const float* sat  = (const float*)d_in[1];   // [N, D]
  const int* lossw  = (const int*)d_in[3];     // beta

  float* distA = (float*)d_ws;                       // 2 - 2*sat@grd^T (rows -> g2s)
  float* distB = distA + (size_t)N_EMB * N_EMB;      // 2 - 2*grd@sat^T (rows -> s2g)
  float* accum = distB + (size_t)N_EMB * N_EMB;      // [2] partial sums

  init_accum_kernel<<<1, 1, 0, stream>>>(accum);

  dim3 grid(N_EMB / BN, N_EMB / BM);
  gemm_dist_kernel<<<grid, THREADS, 0, stream>>>(sat, grd, distA);
  gemm_dist_kernel<<<grid, THREADS, 0, stream>>>(grd, sat, distB);

  topk_softmargin_kernel<<<N_EMB, 256, 0, stream>>>(distA, lossw, accum + 0);
  topk_softmargin_kernel<<<N_EMB, 256, 0, stream>>>(distB, lossw, accum + 1);

  finalize_kernel<<<1, 1, 0, stream>>>(accum, (float*)d_out);
}